// reset_layer_51299089383960
// MI455X (gfx1250) — compile-verified
//
#include <hip/hip_runtime.h>

// Fused block-mixing "reset layer" for MI455X (gfx1250, wave32).
// out[b, jH*200+hc, jW*200+wc] = sum_{iH,iW} R[iH,jH] * L[jW,iW] * x[b, iH*200+hc, iW*200+wc]
//                                + x[b, jH*200+hc, jW*200+wc]
// Two LDS-staged GEMM stages per block using V_WMMA_F32_16X16X4_F32.
// x tile is DMA'd into LDS by the Tensor Data Mover (TENSOR_LOAD_TO_LDS, TENSORcnt).

typedef float v2f __attribute__((ext_vector_type(2)));
typedef float v8f __attribute__((ext_vector_type(8)));
typedef unsigned int u32;
typedef u32 u32x4 __attribute__((ext_vector_type(4)));
typedef u32 u32x8 __attribute__((ext_vector_type(8)));

#define BS     10      // block-mix size
#define CH     200     // chunk length along H and W (2000/10)
#define W_     2000
#define H_     2000
#define WQ     40      // within-chunk columns handled per block
#define NQ     5       // CH / WQ
#define NTOT   400     // BS * WQ  (GEMM N per stage; also xs row stride)
#define KPAD   12      // K padded 10 -> 12 (3 WMMA k-steps of 4)
#define YROW   640     // ys row stride: 16 (padded jW) * WQ
#define NTILES 25      // NTOT / 16

__global__ __launch_bounds__(256)
void reset_fused_wmma(const float* __restrict__ x,
                      const float* __restrict__ Lm,
                      const float* __restrict__ Rm,
                      float* __restrict__ out)
{
    // xs: [iW(12 zero-padded)][iH(10)][wc(40)] -> stage-A B-frag: xs[k*NTOT + n]
    // ys: [iH(12 zero-padded)][jW(16)][wc(40)] -> stage-B B-frag: ys[k*YROW + n]
    __shared__ float xs[KPAD * NTOT];   // 4800 floats
    __shared__ float ys[KPAD * YROW];   // 7680 floats

    const int tid  = threadIdx.x;
    const int lane = tid & 31;
    const int wave = tid >> 5;
    const int l15  = lane & 15;

    const int blk = blockIdx.x;
    const int wq  = blk % NQ;
    const int hc  = (blk / NQ) % CH;
    const int b   = blk / (NQ * CH);

    const float* xb = x   + (size_t)b * H_ * W_;
    float*       ob = out + (size_t)b * H_ * W_;

    // ---- TDM: one descriptor-driven DMA per block fills xs rows 0..9.
    // Tile: X=40 contiguous floats, Y=10 iH rows (stride 200*2000), Z=10 iW segs (stride 200).
    // TDM LDS fill order [Z][Y][X] == xs layout [iW][iH][wc].
    const int uwave = __builtin_amdgcn_readfirstlane(wave);
    if (uwave == 0) {
        unsigned long long ga =
            (unsigned long long)(size_t)(xb + (size_t)hc * W_ + wq * WQ);
        u32x4 g0;
        g0.x = 1u;                                   // count=1 (valid), user mode
        g0.y = (u32)(size_t)&xs[0];                  // lds_addr (bytes)
        g0.z = (u32)(ga & 0xffffffffu);              // global_addr[31:0]
        g0.w = (u32)((ga >> 32) & 0x01ffffffu) | (2u << 30);  // addr[56:32] | type=2
        u32x8 g1;
        g1.s0 = (2u << 16);                          // wg_mask=0, data_size=4B, no opts
        g1.s1 = ((u32)W_ & 0xffffu) << 16;           // tensor_dim0 lo16 (=2000)
        g1.s2 = (((u32)W_ >> 16) & 0xffffu) | (((u32)H_ & 0xffffu) << 16);  // dim0 hi | dim1 lo
        g1.s3 = (((u32)H_ >> 16) & 0xffffu) | ((u32)WQ << 16);              // dim1 hi | tile_dim0=40
        g1.s4 = (u32)BS | ((u32)BS << 16);           // tile_dim1=10 (Y=iH), tile_dim2=10 (Z=iW)
        g1.s5 = (u32)(CH * W_);                      // tensor_dim0_stride lo32 = 400000 (Y)
        g1.s6 = ((u32)CH & 0xffffu) << 16;           // dim0_stride hi16=0 | dim1_stride lo16 = 200 (Z)
        g1.s7 = 0u;                                  // dim1_stride bits 47:16 = 0
        u32x4 g2;
        g2.x = (u32)W_;                              // tensor_dim2 (Z extent, generous)
        g2.y = 1u;                                   // tensor_dim3 (unused; tile_dim3=0)
        g2.z = 0u;                                   // tensor_dim2_stride lo (unused)
        g2.w = 0u;                                   // dim2_stride hi | tile_dim3=0
        asm volatile("tensor_load_to_lds %0, %1, %2, %2"
                     :: "s"(g0), "s"(g1), "s"(g2) : "memory");
    }

    // ---- Zero the K-pad rows (xs rows 10,11; ys rows 10,11) once.
    const float4 z4 = make_float4(0.f, 0.f, 0.f, 0.f);
    for (int i = tid; i < 200; i += 256) *(float4*)&xs[BS * NTOT + i * 4] = z4;
    for (int i = tid; i < 320; i += 256) *(float4*)&ys[BS * YROW + i * 4] = z4;

    // ---- Build A fragments (16x4 f32 layout), branch-free, K padded 10 -> 12.
    // lane<16: M=lane, K = kbase + {0,1}; lane>=16: M=lane-16, K = kbase + {2,3}
    const int mA  = l15;
    const int kh  = (lane >> 4) << 1;         // 0 or 2
    const int mAc = (mA < BS ? mA : BS - 1);  // clamped row for safe load
    v2f aL[3], aR[3];
#pragma unroll
    for (int s = 0; s < 3; ++s) {
        int k0  = s * 4 + kh;
        int k1  = k0 + 1;
        int k0c = (k0 < BS ? k0 : BS - 1);
        int k1c = (k1 < BS ? k1 : BS - 1);
        bool ok0 = (mA < BS) && (k0 < BS);
        bool ok1 = (mA < BS) && (k1 < BS);
        v2f a;
        a.x = ok0 ? Lm[mAc * BS + k0c] : 0.0f;
        a.y = ok1 ? Lm[mAc * BS + k1c] : 0.0f;
        aL[s] = a;
        v2f ar;                                   // A = R^T
        ar.x = ok0 ? Rm[k0c * BS + mAc] : 0.0f;
        ar.y = ok1 ? Rm[k1c * BS + mAc] : 0.0f;
        aR[s] = ar;
    }

    // Issuing wave drains the TDM, then the whole workgroup rendezvouses.
    if (uwave == 0) {
        __builtin_amdgcn_s_wait_tensorcnt((short)0);
    }
    __syncthreads();

    const int mofs = (lane >> 4) * 8;  // C/D row offset for this half-wave

    // ---- Stage A: ys[iH][jW][wc] = sum_iW L[jW,iW] * xs[iW][iH][wc]
    // D rows m>=10 are exactly zero (A rows zeroed), so all 16 rows stored unconditionally.
    for (int t = wave; t < NTILES; t += 8) {
        int n = t * 16 + l15;
        v8f c = {};
#pragma unroll
        for (int s = 0; s < 3; ++s) {
            int k0 = s * 4 + kh;
            v2f bb;
            bb.x = xs[k0 * NTOT + n];
            bb.y = xs[(k0 + 1) * NTOT + n];
            c = __builtin_amdgcn_wmma_f32_16x16x4_f32(
                    false, aL[s], false, bb, (short)0, c, false, false);
        }
        int iHn  = n / WQ;
        int wcn  = n - iHn * WQ;
        int base = iHn * YROW + wcn;
#pragma unroll
        for (int v = 0; v < 8; ++v)
            ys[base + (v + mofs) * WQ] = c[v];
    }

    __syncthreads();

    // ---- Stage B: z[jH] = sum_iH R[iH,jH] * ys[iH][jW][wc]; out = z + identity
    for (int t = wave; t < NTILES; t += 8) {
        int n = t * 16 + l15;
        v8f c = {};
#pragma unroll
        for (int s = 0; s < 3; ++s) {
            int k0 = s * 4 + kh;
            v2f bb;
            bb.x = ys[k0 * YROW + n];
            bb.y = ys[(k0 + 1) * YROW + n];
            c = __builtin_amdgcn_wmma_f32_16x16x4_f32(
                    false, aR[s], false, bb, (short)0, c, false, false);
        }
        int jWn = n / WQ;
        int wcn = n - jWn * WQ;
        int colbase = jWn * CH + wq * WQ + wcn;
        int idbase  = jWn * NTOT + wcn;         // xs identity: (iH=m, iW=jWn, wc)
        // Rows 0,1 (lanes<16) and 8,9 (lanes>=16): full-EXEC stores.
#pragma unroll
        for (int v = 0; v < 2; ++v) {
            int m = v + mofs;
            float idv = xs[idbase + m * WQ];
            ob[(size_t)(m * CH + hc) * W_ + colbase] = c[v] + idv;
        }
        // Rows 2..7 exist only in lanes<16: one masked region, six stores.
        if (lane < 16) {
#pragma unroll
            for (int v = 2; v < 8; ++v) {
                float idv = xs[idbase + v * WQ];
                ob[(size_t)(v * CH + hc) * W_ + colbase] = c[v] + idv;
            }
        }
    }
}

extern "C" void kernel_launch(void* const* d_in, const int* in_sizes, int n_in,
                              void* d_out, int out_size, void* d_ws, size_t ws_size,
                              hipStream_t stream) {
    const float* x  = (const float*)d_in[0];
    const float* Lm = (const float*)d_in[1];
    const float* Rm = (const float*)d_in[2];
    float* out = (float*)d_out;
    dim3 grid(16 * CH * NQ);   // 16000 blocks: (b, hc, wq)
    reset_fused_wmma<<<grid, dim3(256), 0, stream>>>(x, Lm, Rm, out);
}